// GumbelVectorQuantizer_23759759081826
// MI455X (gfx1250) — compile-verified
//
#include <hip/hip_runtime.h>
#include <hip/hip_bf16.h>

// ---------------------------------------------------------------------------
// GumbelVectorQuantizer forward, fused for MI455X (gfx1250, wave32).
//
// Math reduction (see analysis): forward == f32 GEMM logits, then
// idx = argmax_v(logits + gumbel) per (token, group); quantized = codebook
// row gather; avg_probs = histogram(idx)/N.  Memory-bound (~341 MB -> ~15us
// at 23.3 TB/s), so one fused pass; GEMM uses exact-precision
// V_WMMA_F32_16X16X4_F32 to avoid argmax flips from bf16 quantization.
// ---------------------------------------------------------------------------

typedef float v2f __attribute__((ext_vector_type(2)));
typedef float v8f __attribute__((ext_vector_type(8)));

#define N_TOK   32768   // B*T = 16*2048
#define DIM     1024    // d
#define GV      640     // G*V
#define VSZ     320     // V
#define GRP     2       // G
#define DPG     512     // d per group
#define MBLK    32      // tokens per workgroup
#define XS_STR  1028    // LDS row stride (== 4 mod 64 -> conflict-free b64 A reads)

__global__ __launch_bounds__(512)
void gvq_gemm_argmax_gather(const float* __restrict__ x,
                            const float* __restrict__ Wp,
                            const float* __restrict__ bp,
                            const float* __restrict__ cbk,
                            const float* __restrict__ gum,
                            float* __restrict__ outq,
                            int* __restrict__ counts)
{
    extern __shared__ float smem[];
    float* xs   = smem;                      // 32 x 1028 f32 tile of x (131.6 KB)
    float* pv   = xs + MBLK * XS_STR;        // per-wave per-row best value [16][16]
    int*   pc   = (int*)(pv + 256);          // per-wave per-row best column [16][16]
    int*   idxs = pc + 256;                  // final idx per (token,group)  [32][2]

    const int tid     = threadIdx.x;
    const int tokbase = blockIdx.x * MBLK;

    // ---- stage x tile into LDS (float4, coalesced) ----
    const float4* xg = (const float4*)(x + (size_t)tokbase * DIM);
    #pragma unroll
    for (int i = 0; i < 16; ++i) {
        int fid = i * 512 + tid;
        int r = fid >> 8, c4 = fid & 255;          // 256 float4 per row
        float4 v = xg[r * 256 + c4];
        *(float4*)(xs + r * XS_STR + c4 * 4) = v;
    }
    __syncthreads();

    // ---- wave assignment: 16 waves = 2 row-tiles x 8 column-chunks(80 cols) ----
    const int wave  = tid >> 5, lane = tid & 31;
    const int rt    = wave >> 3;                   // row tile (16 tokens)
    const int chunk = wave & 7;                    // 80-column chunk
    const int cb    = chunk * 80;
    const int h     = lane >> 4, m = lane & 15;

    // A fragment: lane m holds row (rt*16+m), K = k0+2h, k0+2h+1  (v2f from LDS)
    const float* arow = xs + (rt * 16 + m) * XS_STR + 2 * h;
    // B fragment: lane holds W[(k0+2h), cb+16t+m] and W[(k0+2h+1), cb+16t+m]
    const float* wbase = Wp + (size_t)(2 * h) * GV + cb + m;

    v8f acc[5] = {};   // 5 tiles of 16x16 f32 C (8 VGPRs each)

    for (int k0 = 0; k0 < DIM; k0 += 4) {
        v2f a = *(const v2f*)(arow + k0);
        const float* wk = wbase + (size_t)k0 * GV;
        #pragma unroll
        for (int t = 0; t < 5; ++t) {
            v2f bb;
            bb.x = wk[t * 16];
            bb.y = wk[t * 16 + GV];
            acc[t] = __builtin_amdgcn_wmma_f32_16x16x4_f32(
                false, a, false, bb, (short)0, acc[t], false, false);
        }
    }

    // ---- fused bias + gumbel + argmax over this wave's 80 columns ----
    float bias[5];
    #pragma unroll
    for (int t = 0; t < 5; ++t) bias[t] = bp[cb + 16 * t + m];

    #pragma unroll
    for (int j = 0; j < 8; ++j) {
        // C layout: lanes 0-15 hold rows j, lanes 16-31 hold rows j+8; col = 16t + (lane&15)
        int rr  = j + 8 * h;
        int tok = tokbase + rt * 16 + rr;
        const float* gp = gum + (size_t)tok * GV;
        float best = -3.4e38f; int bcol = 0;
        #pragma unroll
        for (int t = 0; t < 5; ++t) {
            int col = cb + 16 * t + m;
            float val = acc[t][j] + bias[t] + gp[col];
            if (val > best) { best = val; bcol = col; }   // ascending col: first-max
        }
        // reduce across the 16 lanes of this half-wave (one column per lane)
        #pragma unroll
        for (int s = 8; s >= 1; s >>= 1) {
            float ov = __shfl_xor(best, s, 32);
            int   oc = __shfl_xor(bcol, s, 32);
            if (ov > best || (ov == best && oc < bcol)) { best = ov; bcol = oc; }
        }
        if (m == 0) { pv[wave * 16 + rr] = best; pc[wave * 16 + rr] = bcol; }
    }
    __syncthreads();

    // ---- cross-wave reduce (4 chunks per group), histogram, store idx ----
    if (tid < 64) {
        int R = tid & 31, g = tid >> 5;
        int rtt = R >> 4, r = R & 15;
        float best = -3.4e38f; int bcol = 0x7fffffff;
        #pragma unroll
        for (int c = 0; c < 4; ++c) {
            int w = rtt * 8 + g * 4 + c;
            float v = pv[w * 16 + r];
            int  col = pc[w * 16 + r];
            if (v > best || (v == best && col < bcol)) { best = v; bcol = col; }
        }
        int vsel = bcol - g * VSZ;
        idxs[R * 2 + g] = vsel;
        atomicAdd(&counts[g * VSZ + vsel], 1);
    }
    __syncthreads();

    // ---- gather codebook rows -> quantized output (float4, coalesced) ----
    const float4* cb4 = (const float4*)cbk;
    float4* o4 = (float4*)(outq + (size_t)tokbase * DIM);
    #pragma unroll
    for (int i = 0; i < 16; ++i) {
        int fid = i * 512 + tid;
        int r = fid >> 8, q = fid & 255;        // 256 float4 per token
        int g = q >> 7, k4 = q & 127;           // 128 float4 per group slice
        int vsel = idxs[r * 2 + g];
        o4[r * 256 + q] = cb4[(size_t)(g * VSZ + vsel) * 128 + k4];
    }
}

__global__ void gvq_avg_probs(const int* __restrict__ counts,
                              float* __restrict__ avg)
{
    int i = blockIdx.x * blockDim.x + threadIdx.x;
    if (i < GV) avg[i] = (float)counts[i] * (1.0f / (float)N_TOK);
}

extern "C" void kernel_launch(void* const* d_in, const int* in_sizes, int n_in,
                              void* d_out, int out_size, void* d_ws, size_t ws_size,
                              hipStream_t stream) {
    const float* x    = (const float*)d_in[0];   // (16,2048,1024)
    const float* Wp   = (const float*)d_in[1];   // (1024,640)
    const float* bp   = (const float*)d_in[2];   // (640,)
    const float* cbk  = (const float*)d_in[3];   // (2,320,512)
    const float* gum  = (const float*)d_in[4];   // (32768,2,320)
    float* out = (float*)d_out;                  // quantized (33554432) ++ avg_probs (640)

    int* counts = (int*)d_ws;
    hipMemsetAsync(counts, 0, GV * sizeof(int), stream);

    size_t smem = (size_t)(MBLK * XS_STR + 256 + 256 + 64) * sizeof(float); // ~133.9 KB
    gvq_gemm_argmax_gather<<<N_TOK / MBLK, 512, smem, stream>>>(
        x, Wp, bp, cbk, gum, out, counts);

    gvq_avg_probs<<<1, GV, 0, stream>>>(counts, out + (size_t)N_TOK * DIM);
}